// EvolutionModelTorch_18837726560612
// MI455X (gfx1250) — compile-verified
//
#include <hip/hip_runtime.h>
#include <hip/hip_bf16.h>
#include <math.h>

typedef __attribute__((ext_vector_type(2))) float v2f;
typedef __attribute__((ext_vector_type(4))) float v4f;
typedef __attribute__((ext_vector_type(8))) float v8f;

#define PRIOR_LAMBDA 10.0f

// ---------------------------------------------------------------------------
// Kernel 1: per-batch setup (512 threads total — negligible cost).
// Analytic JC transition matrix: exp(tR) with R = 1/3 offdiag, -1 diag:
//   diag = 1/4 + 3/4 e^{-4t/3},  off = 1/4 - 1/4 e^{-4t/3}
// Fold the per-batch prior scale exp(log_prior/m) into T1.
// Initialize log_p[b] = m * log(0.25) (the constant term of the site log-sum).
// ---------------------------------------------------------------------------
__global__ void jc_setup_kernel(const float* __restrict__ branch1,
                                const float* __restrict__ branch2,
                                float* __restrict__ Tws,   // [B][32] : T1*s | T2
                                float* __restrict__ logp,  // [B]
                                int B, int m) {
  int b = blockIdx.x * blockDim.x + threadIdx.x;
  if (b >= B) return;
  float t1 = branch1[b], t2 = branch2[b];
  float e1 = __expf(-(4.0f / 3.0f) * t1);
  float e2 = __expf(-(4.0f / 3.0f) * t2);
  float dg1 = 0.25f + 0.75f * e1, of1 = 0.25f - 0.25f * e1;
  float dg2 = 0.25f + 0.75f * e2, of2 = 0.25f - 0.25f * e2;
  float s = __expf((__logf(PRIOR_LAMBDA) - PRIOR_LAMBDA * (t1 + t2)) / (float)m);
  float* T = Tws + (size_t)b * 32;
#pragma unroll
  for (int v = 0; v < 4; ++v) {
#pragma unroll
    for (int c = 0; c < 4; ++c) {
      T[v * 4 + c]      = fmaxf((v == c) ? dg1 : of1, 0.0f) * s;
      T[16 + v * 4 + c] = fmaxf((v == c) ? dg2 : of2, 0.0f);
    }
  }
  logp[b] = (float)m * __logf(0.25f);
}

// ---------------------------------------------------------------------------
// Kernel 2: HBM-bound streaming kernel. Each wave32 handles 16-site tiles of
// one batch with V_WMMA_F32_16X16X4_F32 (A = 16x4 feature tile, B = 4x4 T
// padded to 4x16). D is repacked via 256B of wave-private LDS so each of
// lanes 0..15 stores one full site (float4, global_store_b128) and computes
// the row-sum -> log for log_p.
// ---------------------------------------------------------------------------
__launch_bounds__(256)
__global__ void jc_joint_kernel(const float* __restrict__ f1,
                                const float* __restrict__ f2,
                                const float* __restrict__ Tws,
                                float* __restrict__ joint,
                                float* __restrict__ logp,
                                int B, int m, int blocksPerBatch) {
  __shared__ float lds[8 * 64];                 // 256B per wave, 8 waves/block
  const int lane = threadIdx.x & 31;
  const int wid  = threadIdx.x >> 5;
  const int b    = blockIdx.x / blocksPerBatch;
  const int blk  = blockIdx.x % blocksPerBatch;
  const int wavesPerBlock = blockDim.x >> 5;
  const int wavesPerBatch = blocksPerBatch * wavesPerBlock;
  const int wgid  = blk * wavesPerBlock + wid;  // wave index within batch
  const int tiles = m >> 4;                     // 16 sites per tile

  // --- B operands: T matrices (rows K=0..3, cols N=0..3, cols 4..15 zero).
  // 32-bit B 4x16 layout (mirrors C/D row striping):
  //   VGPR0: lanes 0-15 -> K=0, lanes 16-31 -> K=2 ; VGPR1: K=1 / K=3.
  const float* T1 = Tws + (size_t)b * 32;
  const float* T2 = T1 + 16;
  const int n   = lane & 15;
  const int klo = (lane >> 4) << 1;             // 0 or 2
  v2f b1, b2;
  b1.x = (n < 4) ? T1[klo * 4 + n]       : 0.0f;
  b1.y = (n < 4) ? T1[(klo + 1) * 4 + n] : 0.0f;
  b2.x = (n < 4) ? T2[klo * 4 + n]       : 0.0f;
  b2.y = (n < 4) ? T2[(klo + 1) * 4 + n] : 0.0f;

  const size_t batch_off = (size_t)b * (size_t)m * 4;
  const float* F1 = f1 + batch_off;
  const float* F2 = f2 + batch_off;
  float* J = joint + batch_off;
  float* myLds = lds + wid * 64;

  float acc = 0.0f;

  for (int t = wgid; t < tiles; t += wavesPerBatch) {
    const int site0 = t << 4;
    // --- A operand: 32-bit A 16x4 layout:
    //   lane l<16: M=l, VGPR0=K0, VGPR1=K1 ; lane l+16: M=l, K2/K3.
    // => each lane loads one float2; a tile is 256 contiguous bytes.
    const size_t aoff = (size_t)(site0 + n) * 4 + (size_t)((lane >> 4) << 1);
    v2f a1 = *(const v2f*)(F1 + aoff);
    v2f a2 = *(const v2f*)(F2 + aoff);

    // Prefetch the next tile this wave will touch (global_prefetch_b8).
    if (t + wavesPerBatch < tiles) {
      const size_t nxt = aoff + (size_t)wavesPerBatch * 64;
      __builtin_prefetch(F1 + nxt, 0, 1);
      __builtin_prefetch(F2 + nxt, 0, 1);
    }

    v8f cz = {};
    // D = A x B : 16 sites times 4x4 transition matrix.
    v8f d1v = __builtin_amdgcn_wmma_f32_16x16x4_f32(
        false, a1, false, b1, (short)0, cz, false, false);
    v8f d2v = __builtin_amdgcn_wmma_f32_16x16x4_f32(
        false, a2, false, b2, (short)0, cz, false, false);

    v8f p;
#pragma unroll
    for (int g = 0; g < 8; ++g) p[g] = d1v[g] * d2v[g];

    // Repack D through LDS: D VGPR g holds (M=g, N=lane) for lanes 0-15 and
    // (M=g+8, N=lane-16) for lanes 16-31. Only N<4 columns are real.
    if (n < 4) {
      const int siteHi = (lane >> 4) << 3;      // 0 or 8
#pragma unroll
      for (int g = 0; g < 8; ++g)
        myLds[(siteHi + g) * 4 + n] = p[g];
    }
    asm volatile("s_wait_dscnt 0" ::: "memory");

    if (lane < 16) {
      v4f jv = *(const v4f*)(myLds + lane * 4); // one full site per lane
      *(v4f*)(J + (size_t)(site0 + lane) * 4) = jv;   // global_store_b128
      acc += __logf(jv.x + jv.y + jv.z + jv.w);       // site log-sum term
    }
    // LDS ops from the same wave stay in order; next tile's writes cannot
    // pass this tile's reads.
  }

  // Wave reduction of the per-lane log accumulator, one atomic per wave.
#pragma unroll
  for (int off = 16; off >= 1; off >>= 1)
    acc += __shfl_xor(acc, off, 32);
  if (lane == 0) atomicAdd(&logp[b], acc);
}

// ---------------------------------------------------------------------------
extern "C" void kernel_launch(void* const* d_in, const int* in_sizes, int n_in,
                              void* d_out, int out_size, void* d_ws, size_t ws_size,
                              hipStream_t stream) {
  const float* f1  = (const float*)d_in[0];
  const float* br1 = (const float*)d_in[1];
  const float* f2  = (const float*)d_in[2];
  const float* br2 = (const float*)d_in[3];

  const int B = in_sizes[1];                    // 512
  const int m = in_sizes[0] / (B * 4);          // 8192

  float* out   = (float*)d_out;
  float* joint = out;                           // [B, m, 4]
  float* logp  = out + (size_t)B * m * 4;       // [B]
  float* Tws   = (float*)d_ws;                  // [B, 32]

  jc_setup_kernel<<<(B + 255) / 256, 256, 0, stream>>>(br1, br2, Tws, logp, B, m);

  const int blocksPerBatch = 8;                 // 8 blocks x 8 waves = 64 waves/batch
  jc_joint_kernel<<<B * blocksPerBatch, 256, 0, stream>>>(
      f1, f2, Tws, joint, logp, B, m, blocksPerBatch);
}